// CFGSubASTExpressionCombiner_46377056862331
// MI455X (gfx1250) — compile-verified
//
#include <hip/hip_runtime.h>
#include <hip/hip_bf16.h>
#include <math.h>

typedef float v2f __attribute__((ext_vector_type(2)));
typedef float v8f __attribute__((ext_vector_type(8)));

#define HEADS 8
#define DIM   128
#define COMB  1024   // HEADS * DIM

// ---------- helpers ----------

// monotone-increasing f32 -> u32 key (for atomicMax on floats)
__device__ __forceinline__ unsigned f32_key(float x) {
  int i = __float_as_int(x);
  return (unsigned)(i ^ ((i >> 31) | 0x80000000));
}
__device__ __forceinline__ float f32_unkey(unsigned u) {
  int i = (u & 0x80000000u) ? (int)(u ^ 0x80000000u) : ~(int)u;
  return __int_as_float(i);
}

__global__ void zero_f32(float* p, size_t n) {
  size_t i  = (size_t)blockIdx.x * blockDim.x + threadIdx.x;
  size_t st = (size_t)gridDim.x * blockDim.x;
  for (; i < n; i += st) p[i] = 0.0f;
}

// ---------- Wqk[h] = Wq[h] @ Wk[h]^T  (128x128, WMMA f32 16x16x4) ----------
__global__ void wqk_kernel(const float* __restrict__ Wq, const float* __restrict__ Wk,
                           float* __restrict__ Wqk) {
  const int ti = blockIdx.x, tc = blockIdx.y, h = blockIdx.z;
  const int lane = threadIdx.x & 31, r = lane & 15, hf = lane >> 4;
  const int i0 = ti * 16, c0 = tc * 16;
  const float* A  = Wq + (size_t)h * DIM * DIM;   // A[i][j]
  const float* Bk = Wk + (size_t)h * DIM * DIM;   // Wk[c][j]; B[j][c] = Wk[c][j]
  v8f acc = {};
  for (int k = 0; k < DIM; k += 4) {
    v2f a, b;
    a.x = A[(size_t)(i0 + r) * DIM + k + 2 * hf];
    a.y = A[(size_t)(i0 + r) * DIM + k + 2 * hf + 1];
    b.x = Bk[(size_t)(c0 + r) * DIM + k + 2 * hf];
    b.y = Bk[(size_t)(c0 + r) * DIM + k + 2 * hf + 1];
    acc = __builtin_amdgcn_wmma_f32_16x16x4_f32(false, a, false, b, (short)0, acc, false, false);
  }
  float* Dst = Wqk + (size_t)h * DIM * DIM;
  for (int ii = 0; ii < 8; ++ii)
    Dst[(size_t)(i0 + ii + 8 * hf) * DIM + c0 + r] = acc[ii];
}

// ---------- bqk[h][c] = sum_j bq[h][j] * Wk[h][c][j] ----------
__global__ void bqk_kernel(const float* __restrict__ bq, const float* __restrict__ Wk,
                           float* __restrict__ bqk) {
  int idx = blockIdx.x * blockDim.x + threadIdx.x;
  if (idx >= HEADS * DIM) return;
  int h = idx / DIM, c = idx % DIM;
  const float* b = bq + (size_t)h * DIM;
  const float* w = Wk + (size_t)h * DIM * DIM + (size_t)c * DIM;
  float s = 0.0f;
  for (int j = 0; j < DIM; ++j) s += b[j] * w[j];
  bqk[idx] = s;
}

// ---------- Aq[pdg_key[i]] = enc[pdg_value[i]] ----------
__global__ void gather_queries(const float* __restrict__ enc, const int* __restrict__ pdg_key,
                               const int* __restrict__ pdg_value, float* __restrict__ Aq,
                               int n_map) {
  int t = blockIdx.x * blockDim.x + threadIdx.x;
  int i = t >> 5, lane = t & 31;
  if (i >= n_map) return;
  int dst = pdg_key[i], src = pdg_value[i];
  float4 v = ((const float4*)(enc + (size_t)src * DIM))[lane];
  ((float4*)(Aq + (size_t)dst * DIM))[lane] = v;
}

// ---------- S[h] = Aq @ Wqk[h] + bqk[h]  ([N,128] WMMA) ----------
__global__ void s_gemm(const float* __restrict__ Aq, const float* __restrict__ Wqk,
                       const float* __restrict__ bqk, float* __restrict__ S, int N) {
  const int tm = blockIdx.x, h = blockIdx.y;
  const int wave = threadIdx.x >> 5;          // column tile 0..7 (8 waves/block)
  const int lane = threadIdx.x & 31, r = lane & 15, hf = lane >> 4;
  const int m0 = tm * 16, c0 = wave * 16;
  int row = m0 + r; if (row >= N) row = N - 1;   // clamp (no EXEC divergence)
  const float* B = Wqk + (size_t)h * DIM * DIM;
  v8f acc = {};
  for (int k = 0; k < DIM; k += 4) {
    v2f a, b;
    a.x = Aq[(size_t)row * DIM + k + 2 * hf];
    a.y = Aq[(size_t)row * DIM + k + 2 * hf + 1];
    b.x = B[(size_t)(k + 2 * hf) * DIM + c0 + r];
    b.y = B[(size_t)(k + 2 * hf + 1) * DIM + c0 + r];
    acc = __builtin_amdgcn_wmma_f32_16x16x4_f32(false, a, false, b, (short)0, acc, false, false);
  }
  float bias = bqk[h * DIM + c0 + r];
  float* Sh = S + (size_t)h * N * DIM;
  for (int ii = 0; ii < 8; ++ii) {
    int rr = m0 + ii + 8 * hf;
    if (rr < N) Sh[(size_t)rr * DIM + c0 + r] = acc[ii] + bias;
  }
}

// ---------- per-edge scores + segment max (wave per edge) ----------
__global__ void scores_kernel(const float* __restrict__ enc, const int* __restrict__ ast_key,
                              const int* __restrict__ ast_value, const float* __restrict__ S,
                              float* __restrict__ scores, unsigned* __restrict__ segmax,
                              int E, int N) {
  int t = blockIdx.x * blockDim.x + threadIdx.x;
  int e = t >> 5, lane = t & 31;
  if (e >= E) return;                 // wave-uniform
  int src = ast_key[e], seg = ast_value[e];
  float4 v = ((const float4*)(enc + (size_t)src * DIM))[lane];
  const float inv = 0.08838834764831845f;   // 1/sqrt(128)
  for (int h = 0; h < HEADS; ++h) {
    float4 s4 = ((const float4*)(S + ((size_t)h * N + seg) * DIM))[lane];
    float p = v.x * s4.x + v.y * s4.y + v.z * s4.z + v.w * s4.w;
    for (int off = 16; off > 0; off >>= 1) p += __shfl_xor(p, off, 32);
    float sc = p * inv;
    if (lane == 0) {
      scores[(size_t)e * HEADS + h] = sc;
      atomicMax(&segmax[seg * HEADS + h], f32_key(sc));
    }
  }
}

// ---------- exp(score - max), denom accumulation ----------
__global__ void expsum_kernel(const int* __restrict__ ast_value, const unsigned* __restrict__ segmax,
                              float* __restrict__ scores, float* __restrict__ denom, int E) {
  int t = blockIdx.x * blockDim.x + threadIdx.x;
  if (t >= E * HEADS) return;
  int e = t / HEADS, h = t % HEADS;
  int seg = ast_value[e];
  float m  = f32_unkey(segmax[seg * HEADS + h]);
  float ex = __expf(scores[t] - m);
  scores[t] = ex;
  atomicAdd(&denom[seg * HEADS + h], ex);
}

// ---------- weighted scatter: head_acc[seg, h*128:] += p * v (wave per edge) ----------
__global__ void scatter_kernel(const float* __restrict__ enc, const int* __restrict__ ast_key,
                               const int* __restrict__ ast_value, const float* __restrict__ scores,
                               const float* __restrict__ denom, float* __restrict__ head_acc, int E) {
  int t = blockIdx.x * blockDim.x + threadIdx.x;
  int e = t >> 5, lane = t & 31;
  if (e >= E) return;
  int src = ast_key[e], seg = ast_value[e];
  float4 v = ((const float4*)(enc + (size_t)src * DIM))[lane];
  float* base = head_acc + (size_t)seg * COMB;
  for (int h = 0; h < HEADS; ++h) {
    float p = scores[(size_t)e * HEADS + h] / denom[seg * HEADS + h];
    float* b = base + h * DIM + lane * 4;
    atomicAdd(b + 0, p * v.x);
    atomicAdd(b + 1, p * v.y);
    atomicAdd(b + 2, p * v.z);
    atomicAdd(b + 3, p * v.w);
  }
}

// ---------- out = head_acc[N,1024] @ Wo[1024,128] + bo (WMMA) ----------
__global__ void out_gemm(const float* __restrict__ Acc, const float* __restrict__ Wo,
                         const float* __restrict__ bo, float* __restrict__ out, int N) {
  const int tm = blockIdx.x;
  const int wave = threadIdx.x >> 5;          // column tile 0..7
  const int lane = threadIdx.x & 31, r = lane & 15, hf = lane >> 4;
  const int m0 = tm * 16, c0 = wave * 16;
  int row = m0 + r; if (row >= N) row = N - 1;
  v8f acc = {};
  for (int k = 0; k < COMB; k += 4) {
    v2f a, b;
    a.x = Acc[(size_t)row * COMB + k + 2 * hf];
    a.y = Acc[(size_t)row * COMB + k + 2 * hf + 1];
    b.x = Wo[(size_t)(k + 2 * hf) * DIM + c0 + r];
    b.y = Wo[(size_t)(k + 2 * hf + 1) * DIM + c0 + r];
    acc = __builtin_amdgcn_wmma_f32_16x16x4_f32(false, a, false, b, (short)0, acc, false, false);
  }
  float bias = bo[c0 + r];
  for (int ii = 0; ii < 8; ++ii) {
    int rr = m0 + ii + 8 * hf;
    if (rr < N) out[(size_t)rr * DIM + c0 + r] = acc[ii] + bias;
  }
}

// ---------- launcher ----------
extern "C" void kernel_launch(void* const* d_in, const int* in_sizes, int n_in,
                              void* d_out, int out_size, void* d_ws, size_t ws_size,
                              hipStream_t stream) {
  const float* enc = (const float*)d_in[0];
  const float* Wq  = (const float*)d_in[1];
  const float* bq  = (const float*)d_in[2];
  const float* Wk  = (const float*)d_in[3];
  // d_in[4] = bk: per-segment constant in scores -> cancels in segment softmax
  const float* Wo  = (const float*)d_in[5];
  const float* bo  = (const float*)d_in[6];
  const int* ast_key   = (const int*)d_in[7];
  const int* ast_value = (const int*)d_in[8];
  const int* pdg_key   = (const int*)d_in[9];
  const int* pdg_value = (const int*)d_in[10];

  const int E     = in_sizes[7];        // number of edges
  const int n_map = in_sizes[9];        // pdg map entries
  const int N     = out_size / DIM;     // number of CFG nodes (segments)

  // workspace carve (floats)
  float* ws = (float*)d_ws;
  size_t off = 0;
  float* Aq       = ws + off; off += (size_t)N * DIM;
  float* Wqk      = ws + off; off += (size_t)HEADS * DIM * DIM;
  float* bqk      = ws + off; off += (size_t)HEADS * DIM;
  float* S        = ws + off; off += (size_t)HEADS * N * DIM;
  float* scores   = ws + off; off += (size_t)E * HEADS;
  unsigned* segmx = (unsigned*)(ws + off); off += (size_t)N * HEADS;
  float* denom    = ws + off; off += (size_t)N * HEADS;
  float* head_acc = ws + off; off += (size_t)N * COMB;
  (void)ws_size; (void)n_in;

  const int ZB = 4096;
  zero_f32<<<ZB, 256, 0, stream>>>(Aq, (size_t)N * DIM);
  zero_f32<<<ZB, 256, 0, stream>>>((float*)segmx, (size_t)N * HEADS);
  zero_f32<<<ZB, 256, 0, stream>>>(denom, (size_t)N * HEADS);
  zero_f32<<<ZB, 256, 0, stream>>>(head_acc, (size_t)N * COMB);

  wqk_kernel<<<dim3(DIM / 16, DIM / 16, HEADS), 32, 0, stream>>>(Wq, Wk, Wqk);
  bqk_kernel<<<(HEADS * DIM + 255) / 256, 256, 0, stream>>>(bq, Wk, bqk);
  gather_queries<<<(n_map * 32 + 255) / 256, 256, 0, stream>>>(enc, pdg_key, pdg_value, Aq, n_map);
  s_gemm<<<dim3((N + 15) / 16, HEADS), 256, 0, stream>>>(Aq, Wqk, bqk, S, N);
  scores_kernel<<<(E + 7) / 8, 256, 0, stream>>>(enc, ast_key, ast_value, S, scores, segmx, E, N);
  expsum_kernel<<<(E * HEADS + 255) / 256, 256, 0, stream>>>(ast_value, segmx, scores, denom, E);
  scatter_kernel<<<(E + 7) / 8, 256, 0, stream>>>(enc, ast_key, ast_value, scores, denom, head_acc, E);
  out_gemm<<<(N + 15) / 16, 256, 0, stream>>>(head_acc, Wo, bo, (float*)d_out, N);
}